// IIRFilter_6648609374788
// MI455X (gfx1250) — compile-verified
//
#include <hip/hip_runtime.h>
#include <hip/hip_bf16.h>

// ---------------------------------------------------------------------------
// filtfilt (order-8 IIR, K=9 taps) over 1024 rows of T=16384.
// One workgroup (256 threads = 8 wave32) per row; entire row lives in LDS.
// - HBM -> LDS staging:   global_load_async_to_lds_b128  (ASYNCcnt path)
// - IIR:                  chunked state-space scan (64 chunks x 256)
// - init-state fixup:     [256x8]*[8x64] fp32 GEMM via V_WMMA_F32_16X16X4_F32
// - LDS -> HBM store:     global_store_async_from_lds_b32 (reversed, coalesced)
// ---------------------------------------------------------------------------

typedef __attribute__((ext_vector_type(2))) float v2f;
typedef __attribute__((ext_vector_type(8))) float v8f;

#define T_LEN 16384
#define L_CH  256     // chunk length
#define N_CH  64      // chunks per row
#define PITCH 260     // LDS pitch: 260*4 % 16 == 0 -> aligned B128 async writes
#define ORD   8       // IIR order
#define KTAP  9       // FIR taps

// workspace layout (floats):
//  [0..15]   bhat (9 used)
//  [16..31]  ahat (8 used)
//  [32..95]  M = A^256 (8x8 row-major)
//  [128..128+2048) g table: g[t][j] = (e0^T A^{t+1})_j, t=0..255, j=0..7
#define WS_BHAT 0
#define WS_AHAT 16
#define WS_M    32
#define WS_G    128

__device__ __forceinline__ int cidx(int n) { return (n >> 8) * PITCH + (n & 255); }

// -------------------------- setup kernel -----------------------------------
__global__ __launch_bounds__(64) void iir_setup_kernel(const float* __restrict__ b,
                                                       const float* __restrict__ a,
                                                       int K, float* __restrict__ ws)
{
    __shared__ float A[ORD][ORD];
    __shared__ float P[ORD][ORD];
    __shared__ float Pn[ORD][ORD];
    const int tid = threadIdx.x;            // 64 threads
    if (tid == 0) {
        float a0 = a[0];
        for (int k = 0; k < 16; ++k) ws[WS_BHAT + k] = (k < K && k < KTAP) ? b[k] / a0 : 0.0f;
        for (int j = 0; j < 16; ++j) ws[WS_AHAT + j] = (j < ORD && (j + 1) < K) ? a[j + 1] / a0 : 0.0f;
    }
    __syncthreads();
    const int j = tid >> 3, m = tid & 7;
    A[j][m] = (j == 0) ? -ws[WS_AHAT + m] : ((j - 1) == m ? 1.0f : 0.0f);
    __syncthreads();
    P[j][m] = A[j][m];                       // P = A^1
    if (j == 0) ws[WS_G + 0 * ORD + m] = P[0][m];
    __syncthreads();
    for (int t = 1; t < L_CH; ++t) {         // P <- P*A  (P = A^{t+1})
        float acc = 0.0f;
        #pragma unroll
        for (int k = 0; k < ORD; ++k) acc += P[j][k] * A[k][m];
        Pn[j][m] = acc;
        __syncthreads();
        P[j][m] = Pn[j][m];
        if (j == 0) ws[WS_G + t * ORD + m] = acc;
        __syncthreads();
    }
    ws[WS_M + j * ORD + m] = P[j][m];        // M = A^256
}

// -------------------------- one lfilter pass -------------------------------
__device__ __forceinline__ void lfilter_pass(float* __restrict__ f,
                                             const float* __restrict__ in, bool rev,
                                             const float* bb, const float* ah,
                                             const float* __restrict__ gsm,
                                             float* __restrict__ stm,
                                             const float* __restrict__ Mglob, int tid)
{
    // ---- FIR: f[n] = sum_k bhat[k] * x[n-k] (zero history) ----
    for (int n = tid; n < T_LEN; n += 256) {
        float acc = 0.0f;
        #pragma unroll
        for (int k = 0; k < KTAP; ++k) {
            int m = n - k;
            if (m >= 0) {
                int src = rev ? (T_LEN - 1 - m) : m;
                acc += bb[k] * in[cidx(src)];
            }
        }
        f[cidx(n)] = acc;
    }
    __syncthreads();

    // ---- zero-state recurrence, one lane per chunk (in place) ----
    // Dot product explicitly tree-parenthesized: dependent chain ~5 ops/step
    // instead of 9 (compiler cannot reassociate fp).
    if (tid < N_CH) {
        float h0=0,h1=0,h2=0,h3=0,h4=0,h5=0,h6=0,h7=0;
        const int base = tid * PITCH;
        for (int t = 0; t < L_CH; ++t) {
            float p0 = ah[0]*h0 + ah[1]*h1;
            float p1 = ah[2]*h2 + ah[3]*h3;
            float p2 = ah[4]*h4 + ah[5]*h5;
            float p3 = ah[6]*h6 + ah[7]*h7;
            float y  = f[base + t] - ((p0 + p1) + (p2 + p3));
            h7=h6; h6=h5; h5=h4; h4=h3; h3=h2; h2=h1; h1=h0; h0=y;
            f[base + t] = y;
        }
    }
    __syncthreads();

    // ---- chunk-state scan: s_in(c+1) = M*s_in(c) + d_c ----
    if (tid == 0) {
        float s[ORD];
        #pragma unroll
        for (int j = 0; j < ORD; ++j) s[j] = 0.0f;
        float M[ORD * ORD];
        #pragma unroll
        for (int i = 0; i < ORD * ORD; ++i) M[i] = Mglob[i];
        for (int c = 0; c < N_CH; ++c) {
            #pragma unroll
            for (int j = 0; j < ORD; ++j) stm[c * ORD + j] = s[j];  // incoming state
            float ns[ORD];
            #pragma unroll
            for (int j = 0; j < ORD; ++j) {
                const float* Mr = &M[j * ORD];
                float q0 = Mr[0]*s[0] + Mr[1]*s[1];
                float q1 = Mr[2]*s[2] + Mr[3]*s[3];
                float q2 = Mr[4]*s[4] + Mr[5]*s[5];
                float q3 = Mr[6]*s[6] + Mr[7]*s[7];
                ns[j] = f[c * PITCH + (L_CH - 1) - j] + ((q0 + q1) + (q2 + q3));
            }
            #pragma unroll
            for (int j = 0; j < ORD; ++j) s[j] = ns[j];
        }
    }
    __syncthreads();

    // ---- superposition correction: Y[t][c] += g[t][:] . s_in(c)[:] ----
    // [256x8] * [8x64] GEMM via V_WMMA_F32_16X16X4_F32 (two K=4 steps).
    const int lane = tid & 31;
    const int wave = tid >> 5;
    const int rowm = lane & 15;            // A row / B col / D col
    const int kb   = (lane >> 4) << 1;     // K pair selector per half-wave
    for (int tile = wave; tile < 64; tile += 8) {
        const int t0 = (tile >> 2) * 16;
        const int c0 = (tile & 3) * 16;
        v2f a0, a1, b0, b1;
        a0.x = gsm[(t0 + rowm) * ORD + kb];
        a0.y = gsm[(t0 + rowm) * ORD + kb + 1];
        a1.x = gsm[(t0 + rowm) * ORD + 4 + kb];
        a1.y = gsm[(t0 + rowm) * ORD + 4 + kb + 1];
        b0.x = stm[(c0 + rowm) * ORD + kb];
        b0.y = stm[(c0 + rowm) * ORD + kb + 1];
        b1.x = stm[(c0 + rowm) * ORD + 4 + kb];
        b1.y = stm[(c0 + rowm) * ORD + 4 + kb + 1];
        v8f acc = {};
        acc = __builtin_amdgcn_wmma_f32_16x16x4_f32(false, a0, false, b0,
                                                    (short)0, acc, false, false);
        acc = __builtin_amdgcn_wmma_f32_16x16x4_f32(false, a1, false, b1,
                                                    (short)0, acc, false, false);
        const int mbase = t0 + ((lane >> 4) << 3);
        const int cbase = (c0 + rowm) * PITCH;
        #pragma unroll
        for (int v = 0; v < 8; ++v)
            f[cbase + mbase + v] += acc[v];
    }
}

// -------------------------- main kernel ------------------------------------
__global__ __launch_bounds__(256) void filtfilt_kernel(const float* __restrict__ x,
                                                       const float* __restrict__ ws,
                                                       float* __restrict__ out)
{
    extern __shared__ float smem[];
    float* fs  = smem;                      // N_CH*PITCH
    float* fs2 = fs  + N_CH * PITCH;        // N_CH*PITCH
    float* gsm = fs2 + N_CH * PITCH;        // 256*8
    float* stm = gsm + L_CH * ORD;          // 64*8
    float* cf  = stm + N_CH * ORD;          // 32

    const int tid = threadIdx.x;
    const int row = blockIdx.x;
    const float* xrow = x + (size_t)row * T_LEN;
    const unsigned FS2_BYTE = (unsigned)(N_CH * PITCH) * 4u;   // byte offset of fs2 in LDS

    for (int i = tid; i < L_CH * ORD; i += 256) gsm[i] = ws[WS_G + i];
    if (tid < 32) cf[tid] = ws[tid];

    // ---- stage x into fs2 (chunk-major) via async HBM->LDS copies ----
    // Each lane copies 16B/iter; chunk base is 16B aligned (PITCH*4 % 16 == 0).
    for (int i = tid; i < T_LEN / 4; i += 256) {
        const int n = i * 4;                          // 4 floats stay in one chunk
        unsigned           lo = FS2_BYTE + (unsigned)cidx(n) * 4u;
        unsigned long long ga = (unsigned long long)(uintptr_t)(xrow + n);
        asm volatile("global_load_async_to_lds_b128 %0, %1, off"
                     :: "v"(lo), "v"(ga) : "memory");
    }
    asm volatile("s_wait_asynccnt 0x0" ::: "memory");
    __syncthreads();

    float bb[KTAP], ah[ORD];
    #pragma unroll
    for (int k = 0; k < KTAP; ++k) bb[k] = cf[k];
    #pragma unroll
    for (int k = 0; k < ORD; ++k)  ah[k] = cf[16 + k];

    // forward pass: x (fs2) -> y_fwd (fs)
    lfilter_pass(fs, fs2, false, bb, ah, gsm, stm, ws + WS_M, tid);
    __syncthreads();
    // backward pass: reversed y_fwd (fs) -> y_bwd (fs2)
    lfilter_pass(fs2, fs, true, bb, ah, gsm, stm, ws + WS_M, tid);
    __syncthreads();

    // ---- z[n] = y_bwd[T-1-n]: scattered LDS read -> coalesced global write,
    //      via async LDS->HBM stores (S_ENDPGM implicitly waits idle). ----
    const float* orow = out + (size_t)row * T_LEN;
    for (int n = tid; n < T_LEN; n += 256) {
        unsigned           lo = FS2_BYTE + (unsigned)cidx(T_LEN - 1 - n) * 4u;
        unsigned long long ga = (unsigned long long)(uintptr_t)(orow + n);
        asm volatile("global_store_async_from_lds_b32 %0, %1, off"
                     :: "v"(ga), "v"(lo) : "memory");
    }
    asm volatile("s_wait_asynccnt 0x0" ::: "memory");
}

// -------------------------- launcher ---------------------------------------
extern "C" void kernel_launch(void* const* d_in, const int* in_sizes, int n_in,
                              void* d_out, int out_size, void* d_ws, size_t ws_size,
                              hipStream_t stream) {
    const float* x = (const float*)d_in[0];
    const float* b = (const float*)d_in[1];
    const float* a = (const float*)d_in[2];
    float* out = (float*)d_out;
    float* ws  = (float*)d_ws;
    const int K    = in_sizes[1];                 // 9
    const int rows = in_sizes[0] / T_LEN;         // 1024

    iir_setup_kernel<<<1, 64, 0, stream>>>(b, a, K, ws);

    const size_t shbytes =
        (size_t)(2 * N_CH * PITCH + L_CH * ORD + N_CH * ORD + 32) * sizeof(float);
    (void)hipFuncSetAttribute((const void*)filtfilt_kernel,
                              hipFuncAttributeMaxDynamicSharedMemorySize,
                              (int)shbytes);
    filtfilt_kernel<<<rows, 256, shbytes, stream>>>(x, ws, out);
}